// YOLOLoss_63075889709485
// MI455X (gfx1250) — compile-verified
//
#include <hip/hip_runtime.h>
#include <math.h>

// ---------------------------------------------------------------------------
// YOLOv3-style loss for MI455X (gfx1250, wave32).
// Memory-bound (~116 MB in -> ~5us roofline @ 23.3 TB/s). Optimizations:
//  * obj-sparse branch: obj==0 cells (>99.7%) touch only 6 of 170 floats
//  * target boxes staged to LDS once per block via the Tensor Data Mover
//    (tensor_load_to_lds + s_wait_tensorcnt), not per-lane loads
//  * WMMA f32 16x16x4 ones-matrix trick for the wave32 reduction
//  * ds_swizzle SWAPX16 for the final cross-half-wave add
// ---------------------------------------------------------------------------

typedef __attribute__((ext_vector_type(2))) float    v2f;
typedef __attribute__((ext_vector_type(8))) float    v8f;
typedef __attribute__((ext_vector_type(4))) unsigned v4u;
typedef __attribute__((ext_vector_type(8))) unsigned v8u;

#define NBOX 32
#define BATCH 16
#define BLOCK 256

__device__ __forceinline__ float bce(float p, float t) {
    // -(t*log(p) + (1-t)*log(1-p)); p is guaranteed in (eps, 1-eps)
    return -(t * logf(p) + (1.0f - t) * logf(1.0f - p));
}

// Wave32 reduction via V_WMMA_F32_16X16X4_F32 with B = ones.
// A (16x4 f32): VGPR0 -> K=0 (lanes 0-15) / K=2 (lanes 16-31); VGPR1 -> K=1/K=3.
// With a1=0: D[m][n] = p(m) + p(m+16) for all n. Each lane sums its 8 D regs
// (rows 0-7 or 8-15 of its column), then one SWAPX16 ds_swizzle pairs lane l
// with l^16 so every lane holds the full 32-lane sum. EXEC must be all ones.
__device__ __forceinline__ float wave_reduce_wmma(float p) {
    v2f a; a[0] = p;    a[1] = 0.0f;
    v2f b; b[0] = 1.0f; b[1] = 1.0f;
    v8f c = {};
    v8f d = __builtin_amdgcn_wmma_f32_16x16x4_f32(
        /*neg_a=*/false, a, /*neg_b=*/false, b,
        /*c_mod=*/(short)0, c, /*reuse_a=*/false, /*reuse_b=*/false);
    float s = d[0] + d[1] + d[2] + d[3] + d[4] + d[5] + d[6] + d[7];
    // SWAPX16: group-of-32 swizzle, xor_mask=0x10, and_mask=0x1f -> 0x401F
    s += __int_as_float(__builtin_amdgcn_ds_swizzle(__float_as_int(s), 0x401F));
    return s;
}

__global__ void init_out(float* out) { out[0] = 0.0f; }

// One thread per batch image: collect boxes of cells with obj==1 (<=20 < 32),
// in increasing flat index order (equivalent to top_k of a 0/1 vector + valid
// masking: all ones are selected, zero entries contribute iou=0).
__global__ void gather_boxes(const float* __restrict__ tgt,
                             float4* __restrict__ boxes,
                             int* __restrict__ counts, int g) {
    int bimg = threadIdx.x;
    if (bimg >= BATCH) return;
    const int n = g * g * 3;
    const float* tb = tgt + (size_t)bimg * n * 85;
    float4* ob = boxes + bimg * NBOX;
    int cnt = 0;
    for (int i = 0; i < n && cnt < NBOX; ++i) {
        const float* cell = tb + (size_t)i * 85;
        if (cell[4] > 0.5f) {
            ob[cnt] = make_float4(cell[0], cell[1], cell[2], cell[3]);
            ++cnt;
        }
    }
    counts[bimg] = cnt;
}

// TDM copy: 512B (128 x 4B elements, 1-D tile) global -> LDS, single DMA.
// D# group0: count=1, lds_addr, 57-bit global_addr, type=2 ("image").
// D# group1: data_size=2 (4B), tensor_dim0=tile_dim0=128, dim1=tile_dim1=1,
//            tensor_dim0_stride=128, no padding/iterate/multicast.
__device__ __forceinline__ void tdm_load_boxes_to_lds(const void* gsrc,
                                                      void* lds_dst) {
    const unsigned long long ga = (unsigned long long)(uintptr_t)gsrc;
    const unsigned lds = (unsigned)(uintptr_t)lds_dst;  // low 32 bits = LDS offset
    v4u g0;
    g0[0] = 1u;                                    // count=1, user mode
    g0[1] = lds;                                   // lds_addr
    g0[2] = (unsigned)(ga & 0xFFFFFFFFu);          // global_addr[31:0]
    g0[3] = (unsigned)((ga >> 32) & 0x01FFFFFFu)   // global_addr[56:32]
          | (2u << 30);                            // type=2
    v8u g1;
    g1[0] = (2u << 16);                            // data_size=4B
    g1[1] = (128u << 16);                          // tensor_dim0[15:0]=128
    g1[2] = (1u << 16);                            // tensor_dim1=1
    g1[3] = (128u << 16);                          // tile_dim0=128
    g1[4] = 1u;                                    // tile_dim1=1, tile_dim2=0
    g1[5] = 128u;                                  // tensor_dim0_stride=128
    g1[6] = 0u;
    g1[7] = 0u;
    asm volatile("tensor_load_to_lds %0, %1" :: "s"(g0), "s"(g1) : "memory");
    __builtin_amdgcn_s_wait_tensorcnt(0);
}

__global__ __launch_bounds__(BLOCK)
void yolo_layer_loss(const float* __restrict__ pred,
                     const float* __restrict__ tgt,
                     const float4* __restrict__ boxes,
                     const int* __restrict__ counts,
                     float* __restrict__ out, int g,
                     float aw0, float ah0, float aw1, float ah1,
                     float aw2, float ah2) {
    __shared__ float4 sbox[NBOX];
    __shared__ int   scnt;
    __shared__ float wsum[BLOCK / 32];

    const int bimg = blockIdx.y;
    const int tid  = threadIdx.x;

    if (tid < 32) {  // wave 0 issues one TDM DMA (EXEC ignored by tensor ops)
        tdm_load_boxes_to_lds(boxes + bimg * NBOX, sbox);
    }
    if (tid == 0) scnt = counts[bimg];
    __syncthreads();

    const int n = g * g * 3;
    const int i = blockIdx.x * BLOCK + tid;

    float loss = 0.0f;
    if (i < n) {
        const int a  = i % 3;
        const int xi = (i / 3) % g;
        const int yi = i / (3 * g);
        const size_t base = ((size_t)bimg * n + i) * 85;
        const float* P = pred + base;
        const float* T = tgt  + base;

        const float gw = (float)g;
        const float gx = (float)xi, gy = (float)yi;
        const float aw = (a == 0) ? aw0 : ((a == 1) ? aw1 : aw2);
        const float ah = (a == 0) ? ah0 : ((a == 1) ? ah1 : ah2);

        const float obj = T[4];
        const float p4  = P[4];
        const float p0 = P[0], p1 = P[1], p2 = P[2], p3 = P[3];

        if (obj > 0.5f) {
            // full-channel path (~0.25% of cells): prefetch the class run
            __builtin_prefetch(&P[5], 0, 0);
            __builtin_prefetch(&T[5], 0, 0);
            const float t0 = T[0], t1 = T[1], t2 = T[2], t3 = T[3];
            const float scale = 2.0f - t2 * t3;
            const float rtx = t0 * gw - gx;
            const float rty = t1 * gw - gy;
            const float rtw = logf(t2 * 416.0f / aw + 1e-12f);
            const float rth = logf(t3 * 416.0f / ah + 1e-12f);
            loss += scale * (bce(p0, rtx) + bce(p1, rty));
            const float dw = p2 - rtw, dh = p3 - rth;
            loss += scale * (dw * dw + dh * dh);
            loss += -logf(p4);                    // bce(p4, 1) * 1
            #pragma unroll 5
            for (int ch = 5; ch < 85; ++ch)       // class BCE, obj==1 only
                loss += bce(P[ch], T[ch]);
        } else {
            // obj==0: only the ignore-masked confidence term matters
            const float px = (p0 + gx) / gw;
            const float py = (p1 + gy) / gw;
            const float pw = expf(p2) * aw * (1.0f / 416.0f);
            const float ph = expf(p3) * ah * (1.0f / 416.0f);
            const float phw = 0.5f * pw, phh = 0.5f * ph;
            const float pminx = px - phw, pmaxx = px + phw;
            const float pminy = py - phh, pmaxy = py + phh;
            const float parea = pw * ph;
            float maxiou = 0.0f;
            const int cnt = scnt;
            for (int k = 0; k < cnt; ++k) {
                const float4 bb = sbox[k];
                const float bhw = 0.5f * bb.z, bhh = 0.5f * bb.w;
                const float ix = fminf(pmaxx, bb.x + bhw) - fmaxf(pminx, bb.x - bhw);
                const float iy = fminf(pmaxy, bb.y + bhh) - fmaxf(pminy, bb.y - bhh);
                const float inter = fmaxf(ix, 0.0f) * fmaxf(iy, 0.0f);
                const float iou = inter / (parea + bb.z * bb.w - inter);
                maxiou = fmaxf(maxiou, iou);
            }
            if (maxiou < 0.5f) loss = -logf(1.0f - p4);  // bce(p4,0)*ignore
        }
    }

    // ---- block reduction (EXEC is full here: no early returns above) ----
    const float s = wave_reduce_wmma(loss);
    if ((tid & 31) == 0) wsum[tid >> 5] = s;
    __syncthreads();
    if (tid == 0) {
        float t = 0.0f;
        #pragma unroll
        for (int w = 0; w < BLOCK / 32; ++w) t += wsum[w];
        atomicAdd(out, t * (1.0f / (float)BATCH));   // mean over batch
    }
}

extern "C" void kernel_launch(void* const* d_in, const int* in_sizes, int n_in,
                              void* d_out, int out_size, void* d_ws, size_t ws_size,
                              hipStream_t stream) {
    // setup_inputs() dict order: preds0, targets0, preds1, targets1, preds2, targets2
    const float* preds[3] = {(const float*)d_in[0], (const float*)d_in[2], (const float*)d_in[4]};
    const float* tgts [3] = {(const float*)d_in[1], (const float*)d_in[3], (const float*)d_in[5]};
    float* out = (float*)d_out;

    float4* boxes = (float4*)d_ws;                                 // 3*16*32 float4
    int*    counts = (int*)((char*)d_ws + 3 * BATCH * NBOX * sizeof(float4));

    const int grids[3] = {13, 26, 52};
    // ANCHORS[ANCHOR_MASK[l]]
    const float anc[3][6] = {
        {116.0f, 90.0f, 156.0f, 198.0f, 373.0f, 326.0f},   // layer 0 (13x13)
        { 30.0f, 61.0f,  62.0f,  45.0f,  59.0f, 119.0f},   // layer 1 (26x26)
        { 10.0f, 13.0f,  16.0f,  30.0f,  33.0f,  23.0f},   // layer 2 (52x52)
    };

    init_out<<<1, 1, 0, stream>>>(out);
    for (int l = 0; l < 3; ++l) {
        gather_boxes<<<1, BATCH, 0, stream>>>(tgts[l], boxes + l * BATCH * NBOX,
                                              counts + l * BATCH, grids[l]);
    }
    for (int l = 0; l < 3; ++l) {
        const int n = grids[l] * grids[l] * 3;
        dim3 grid((n + BLOCK - 1) / BLOCK, BATCH);
        yolo_layer_loss<<<grid, BLOCK, 0, stream>>>(
            preds[l], tgts[l], boxes + l * BATCH * NBOX, counts + l * BATCH, out,
            grids[l], anc[l][0], anc[l][1], anc[l][2], anc[l][3], anc[l][4], anc[l][5]);
    }
}